// VVGRAPHmodel_34548716929569
// MI455X (gfx1250) — compile-verified
//
#include <hip/hip_runtime.h>
#include <hip/hip_bf16.h>

// CDNA5 / gfx1250: wave32, WMMA (not MFMA). V_WMMA_F32_16X16X4_F32 for all
// dense node-wise linears; f32 global atomics for segment-mean aggregation.

typedef float v2f __attribute__((ext_vector_type(2)));
typedef float v8f __attribute__((ext_vector_type(8)));

#define N_IN 15

__device__ __forceinline__ int imin(int a, int b) { return a < b ? a : b; }

// ---------------------------------------------------------------------------
// Fused WMMA linear:
//   out[r, :NOUT] = act( scale1(A1[r,:K]) @ W1 (+ A2[r,:K] @ W2) + bias
//                        (+ madd[r,:]/max(deg[r],1)) )
// scale1 = 1/max(deg[r],1) when MEAN_A1 (SAGE mean-agg folded into A-frags).
//
// Branch-free inner loop: A/B tiles staged zero-padded in LDS (clamped
// unconditional global loads + select), fragments read as aligned b64 LDS
// loads, epilogue goes through LDS for coalesced global stores.
// One wave -> 16 rows x NT 16-wide column tiles; 8 waves/block -> 128 rows/block.
// ---------------------------------------------------------------------------
template<int K, int NOUT, bool HAS_A2, bool MEAN_A1, bool HAS_MADD, bool RELU>
__global__ __launch_bounds__(256)
void wmma_linear(const float* __restrict__ A1,
                 const float* __restrict__ W1g,
                 const float* __restrict__ A2,
                 const float* __restrict__ W2g,
                 const float* __restrict__ biasg,
                 const float* __restrict__ madd,
                 const float* __restrict__ deg,
                 float* __restrict__ out, int nrows) {
    constexpr int KT = (K + 3) / 4;       // k-tiles of 4
    constexpr int NT = (NOUT + 15) / 16;  // 16-wide column tiles

    __shared__ __align__(16) float As[128 * 32];
    __shared__ __align__(16) float A2s[HAS_A2 ? 128 * 32 : 8];
    __shared__ __align__(16) float Bs[KT * 128];   // [kk][hi][col][2]
    __shared__ __align__(16) float B2s[HAS_A2 ? KT * 128 : 8];
    __shared__ __align__(16) float Cs[128 * 32];
    __shared__ float bias_s[32];

    const int tid      = threadIdx.x;
    const int rowBase  = blockIdx.x * 128;

    // ---- stage B (fragment-major: pair (k_even,k_odd) contiguous) ----
    for (int j = tid; j < KT * 128; j += 256) {
        const int p  = j & 1;
        const int c  = (j >> 1) & 31;
        const int kh = j >> 6;            // kk*2 + hi
        const int k  = kh * 2 + p;
        const bool ok = (k < K) && (c < NOUT);
        const int  ai = imin(k, K - 1) * NOUT + imin(c, NOUT - 1);
        Bs[j] = ok ? W1g[ai] : 0.0f;
        if constexpr (HAS_A2) B2s[j] = ok ? W2g[ai] : 0.0f;
    }
    if (tid < 32) {
        const float bv = biasg[imin(tid, NOUT - 1)];
        bias_s[tid] = (tid < NOUT) ? bv : 0.0f;
    }

    // ---- stage A tile(s), zero-padded [128][32], coalesced ----
    for (int j = tid; j < 128 * 32; j += 256) {
        const int k  = j & 31;
        const int rl = j >> 5;
        const int r  = rowBase + rl;
        const bool ok = (r < nrows) && (k < K);
        const int  ai = imin(r, nrows - 1) * K + imin(k, K - 1);
        As[j] = ok ? A1[ai] : 0.0f;
        if constexpr (HAS_A2) A2s[j] = ok ? A2[ai] : 0.0f;
    }
    __syncthreads();

    const int lane = tid & 31;
    const int wv   = tid >> 5;
    const int lrow = lane & 15;
    const int hi   = lane >> 4;           // 0: K=0,1  1: K=2,3
    const int arl  = wv * 16 + lrow;      // block-local A row for this lane

    float sc1 = 1.0f;
    if constexpr (MEAN_A1) {
        const int gr = imin(rowBase + arl, nrows - 1);
        sc1 = 1.0f / fmaxf(deg[gr], 1.0f);
    }

    v8f acc[NT];
#pragma unroll
    for (int nt = 0; nt < NT; ++nt)
        acc[nt] = (v8f){0.f, 0.f, 0.f, 0.f, 0.f, 0.f, 0.f, 0.f};

#pragma unroll
    for (int kk = 0; kk < KT; ++kk) {
        const int kb = kk * 4 + hi * 2;
        v2f a = *(const v2f*)&As[arl * 32 + kb];
        if constexpr (MEAN_A1) a = a * sc1;
#pragma unroll
        for (int nt = 0; nt < NT; ++nt) {
            const v2f b = *(const v2f*)&Bs[((kk * 2 + hi) * 32 + nt * 16 + lrow) * 2];
            acc[nt] = __builtin_amdgcn_wmma_f32_16x16x4_f32(
                false, a, false, b, (short)0, acc[nt], false, false);
        }
        if constexpr (HAS_A2) {
            const v2f a2 = *(const v2f*)&A2s[arl * 32 + kb];
#pragma unroll
            for (int nt = 0; nt < NT; ++nt) {
                const v2f b2 = *(const v2f*)&B2s[((kk * 2 + hi) * 32 + nt * 16 + lrow) * 2];
                acc[nt] = __builtin_amdgcn_wmma_f32_16x16x4_f32(
                    false, a2, false, b2, (short)0, acc[nt], false, false);
            }
        }
    }

    // ---- C tiles -> LDS (layout: VGPR j: lanes0-15 M=j, lanes16-31 M=j+8) ----
#pragma unroll
    for (int nt = 0; nt < NT; ++nt) {
#pragma unroll
        for (int j = 0; j < 8; ++j) {
            const int rl = wv * 16 + (hi ? j + 8 : j);
            Cs[rl * 32 + nt * 16 + lrow] = acc[nt][j];
        }
    }
    __syncthreads();

    // ---- cooperative, coalesced epilogue ----
    for (int j = tid; j < 128 * NOUT; j += 256) {
        const int c  = j % NOUT;          // compile-time-constant modulus
        const int rl = j / NOUT;
        const int r  = rowBase + rl;
        if (r < nrows) {
            float v = Cs[rl * 32 + c] + bias_s[c];
            if constexpr (HAS_MADD)
                v += madd[r * NOUT + c] * (1.0f / fmaxf(deg[r], 1.0f));
            if constexpr (RELU) v = fmaxf(v, 0.0f);
            out[r * NOUT + c] = v;
        }
    }
}

// ---------------------------------------------------------------------------
// NNConv1 edge kernel: h = relu(attr@w1+b1) (2->12); per-edge weight matrix
// never materialized: msg[o] = sum_i x[src,i] * (b2 + h@w2)[i*18+o].
// w2 kept K-transposed in LDS so the 12-wide dot is contiguous + broadcast.
// Also accumulates deg[dst] (reused by every layer's segment-mean).
// ---------------------------------------------------------------------------
__global__ __launch_bounds__(256)
void nn1_edge(const float* __restrict__ x,
              const int* __restrict__ src, const int* __restrict__ dst,
              const float* __restrict__ attr,
              const float* __restrict__ w1, const float* __restrict__ b1,
              const float* __restrict__ w2, const float* __restrict__ b2,
              float* __restrict__ agg, float* __restrict__ deg, int E) {
    __shared__ float w1s[24];           // [2][12]
    __shared__ float b1s[12];
    __shared__ __align__(16) float w2t[270 * 12];   // transposed: [i*18+o][k]
    __shared__ float b2s[270];

    const int tid = threadIdx.x;
    if (tid < 24) w1s[tid] = w1[tid];
    if (tid < 12) b1s[tid] = b1[tid];
    for (int i = tid; i < 3240; i += 256) {
        const int k = i / 270, io = i % 270;
        w2t[io * 12 + k] = w2[i];
    }
    for (int i = tid; i < 270; i += 256) b2s[i] = b2[i];
    __syncthreads();

    const int e = blockIdx.x * 256 + tid;
    if (e >= E) return;

    const float a0 = attr[2 * e], a1 = attr[2 * e + 1];
    float h[12];
#pragma unroll
    for (int k = 0; k < 12; ++k)
        h[k] = fmaxf(fmaf(a1, w1s[12 + k], fmaf(a0, w1s[k], b1s[k])), 0.0f);

    const int s = src[e], d = dst[e];
    const float* xr = x + s * N_IN;
    float msg[18];
#pragma unroll
    for (int o = 0; o < 18; ++o) msg[o] = 0.0f;

    for (int i = 0; i < N_IN; ++i) {
        const float xi = xr[i];
        const int base = i * 18;
        for (int o = 0; o < 18; ++o) {
            const float* wp = &w2t[(base + o) * 12];
            float wvv = b2s[base + o];
#pragma unroll
            for (int k = 0; k < 12; ++k) wvv = fmaf(h[k], wp[k], wvv);
            msg[o] = fmaf(xi, wvv, msg[o]);
        }
    }
#pragma unroll
    for (int o = 0; o < 18; ++o) atomicAdd(&agg[d * 18 + o], msg[o]);
    atomicAdd(&deg[d], 1.0f);
}

// ---------------------------------------------------------------------------
// SAGE scatter: agg[dst,f] += hp[src,f]; one thread per (edge, feature).
// ---------------------------------------------------------------------------
template<int D>
__global__ __launch_bounds__(256)
void scatter_add(const float* __restrict__ hp,
                 const int* __restrict__ src, const int* __restrict__ dst,
                 float* __restrict__ agg, int total) {
    const int idx = blockIdx.x * 256 + threadIdx.x;
    if (idx >= total) return;
    const int e = idx / D;
    const int f = idx - e * D;
    atomicAdd(&agg[dst[e] * D + f], hp[src[e] * D + f]);
}

// ---------------------------------------------------------------------------
// NNConv2 edge kernel (30 -> 1): scalar message per edge.
// ---------------------------------------------------------------------------
__global__ __launch_bounds__(256)
void nn2_edge(const float* __restrict__ h4,
              const int* __restrict__ src, const int* __restrict__ dst,
              const float* __restrict__ attr,
              const float* __restrict__ w1, const float* __restrict__ b1,
              const float* __restrict__ w2, const float* __restrict__ b2,
              float* __restrict__ agg, int E) {
    __shared__ float w1s[16], b1s[8], w2s[240], b2s[30];
    const int tid = threadIdx.x;
    if (tid < 16)  w1s[tid] = w1[tid];
    if (tid < 8)   b1s[tid] = b1[tid];
    if (tid < 240) w2s[tid] = w2[tid];
    if (tid < 30)  b2s[tid] = b2[tid];
    __syncthreads();

    const int e = blockIdx.x * 256 + tid;
    if (e >= E) return;

    const float a0 = attr[2 * e], a1 = attr[2 * e + 1];
    float h[8];
#pragma unroll
    for (int k = 0; k < 8; ++k)
        h[k] = fmaxf(fmaf(a1, w1s[8 + k], fmaf(a0, w1s[k], b1s[k])), 0.0f);

    const int s = src[e], d = dst[e];
    const float* hr = h4 + s * 30;
    float msg = 0.0f;
    for (int i = 0; i < 30; ++i) {
        float wvv = b2s[i];
#pragma unroll
        for (int k = 0; k < 8; ++k) wvv = fmaf(h[k], w2s[k * 30 + i], wvv);
        msg = fmaf(hr[i], wvv, msg);
    }
    atomicAdd(&agg[d], msg);
}

// ---------------------------------------------------------------------------
// Final: out = relu(x[:,1] + (agg/deg + h4@root + bias) + avg_g)
// ---------------------------------------------------------------------------
__global__ __launch_bounds__(256)
void final_node(const float* __restrict__ x, const float* __restrict__ h4,
                const float* __restrict__ agg, const float* __restrict__ deg,
                const float* __restrict__ avg_g,
                const float* __restrict__ root2, const float* __restrict__ bias2,
                float* __restrict__ out, int n) {
    const int i = blockIdx.x * 256 + threadIdx.x;
    if (i >= n) return;
    const float inv = 1.0f / fmaxf(deg[i], 1.0f);
    float acc = fmaf(agg[i], inv, bias2[0]);
    const float* hr = h4 + i * 30;
#pragma unroll
    for (int k = 0; k < 30; ++k) acc = fmaf(hr[k], root2[k], acc);
    out[i] = fmaxf(x[i * N_IN + 1] + acc + avg_g[i], 0.0f);
}

// ---------------------------------------------------------------------------
extern "C" void kernel_launch(void* const* d_in, const int* in_sizes, int n_in,
                              void* d_out, int out_size, void* d_ws, size_t ws_size,
                              hipStream_t stream) {
    const int n = in_sizes[0] / N_IN;   // 100000
    const int E = in_sizes[2] / 2;      // 1000000

    const float* x     = (const float*)d_in[0];
    const int*   ei    = (const int*)  d_in[1];
    const float* attr  = (const float*)d_in[2];
    const float* avg_g = (const float*)d_in[3];
    // params flattened in sorted-key pytree order:
    const float* n1_b1 = (const float*)d_in[4];
    const float* n1_b2 = (const float*)d_in[5];
    const float* n1_bi = (const float*)d_in[6];
    const float* n1_rt = (const float*)d_in[7];
    const float* n1_w1 = (const float*)d_in[8];
    const float* n1_w2 = (const float*)d_in[9];
    const float* n2_b1 = (const float*)d_in[10];
    const float* n2_b2 = (const float*)d_in[11];
    const float* n2_bi = (const float*)d_in[12];
    const float* n2_rt = (const float*)d_in[13];
    const float* n2_w1 = (const float*)d_in[14];
    const float* n2_w2 = (const float*)d_in[15];
    // sage L: lb, lw, proj_b, proj_w, rw starting at 16 + 5*L
    const int* src = ei;
    const int* dst = ei + E;

    float* ws  = (float*)d_ws;
    const size_t N = (size_t)n;
    float* deg = ws;
    float* agg = ws + N;             // N*27 max, re-zeroed per layer
    float* hp  = ws + N * 28;        // N*30
    float* h0  = ws + N * 58;        // N*18
    float* h1  = ws + N * 76;        // N*21
    float* h2  = ws + N * 97;        // N*24
    float* h3  = ws + N * 121;       // N*27
    float* h4  = ws + N * 148;       // N*30

    const int EB = (E + 255) / 256;
    const int GB = (n + 127) / 128;  // 128 rows / block in wmma_linear
    const int NB = (n + 255) / 256;

    // ---- NNConv1 ----
    hipMemsetAsync(deg, 0, N * sizeof(float), stream);
    hipMemsetAsync(agg, 0, N * 18 * sizeof(float), stream);
    nn1_edge<<<EB, 256, 0, stream>>>(x, src, dst, attr, n1_w1, n1_b1, n1_w2, n1_b2, agg, deg, E);
    // h0 = relu(x@root + agg/deg + bias)
    wmma_linear<15, 18, false, false, true, true><<<GB, 256, 0, stream>>>(
        x, n1_rt, nullptr, nullptr, n1_bi, agg, deg, h0, n);

    // ---- SAGE layers ----
#define SAGE_LAYER(L, DIN, DOUT, HIN, HOUT)                                             \
    {                                                                                   \
        const float* lb = (const float*)d_in[16 + 5 * (L)];                             \
        const float* lw = (const float*)d_in[17 + 5 * (L)];                             \
        const float* pb = (const float*)d_in[18 + 5 * (L)];                             \
        const float* pw = (const float*)d_in[19 + 5 * (L)];                             \
        const float* rw = (const float*)d_in[20 + 5 * (L)];                             \
        wmma_linear<DIN, DIN, false, false, false, true><<<GB, 256, 0, stream>>>(       \
            HIN, pw, nullptr, nullptr, pb, nullptr, nullptr, hp, n);                    \
        hipMemsetAsync(agg, 0, N * (DIN) * sizeof(float), stream);                      \
        scatter_add<DIN><<<((E * (DIN)) + 255) / 256, 256, 0, stream>>>(                \
            hp, src, dst, agg, E * (DIN));                                              \
        wmma_linear<DIN, DOUT, true, true, false, true><<<GB, 256, 0, stream>>>(        \
            agg, lw, HIN, rw, lb, nullptr, deg, HOUT, n);                               \
    }

    SAGE_LAYER(0, 18, 21, h0, h1)
    SAGE_LAYER(1, 21, 24, h1, h2)
    SAGE_LAYER(2, 24, 27, h2, h3)
    SAGE_LAYER(3, 27, 30, h3, h4)
#undef SAGE_LAYER

    // ---- NNConv2 + final ----
    hipMemsetAsync(agg, 0, N * sizeof(float), stream);
    nn2_edge<<<EB, 256, 0, stream>>>(h4, src, dst, attr, n2_w1, n2_b1, n2_w2, n2_b2, agg, E);
    final_node<<<NB, 256, 0, stream>>>(x, h4, agg, deg, avg_g, n2_rt, n2_bi,
                                       (float*)d_out, n);
}